// PCEN_27101243638438
// MI455X (gfx1250) — compile-verified
//
#include <hip/hip_runtime.h>
#include <math.h>

// ---------------------------------------------------------------------------
// PCEN for (B=8, T=16384, C=128) f32 on MI455X (gfx1250, wave32).
//
// EMA scan y[t] = s*x[t] + q*y[t-1] recast per 16-timestep tile as
//   Y(16x16) = A(16x16 lower-tri const) * V(16 time x 16 chan) + p (x) carry
// via 4 chained V_WMMA_F32_16X16X4_F32 (K=16), carry folded into the initial
// C accumulator; new carry = Y row 15 broadcast with ds_bpermute.
//
// T split into 32 partitions of 512 steps; partitions p>0 warm up over a
// 512-step halo with carry=0 (q^512 ~ 9e-10, below fp32 eps) -> 2048 fully
// independent waves (8 B x 8 ctiles x 32 parts). x (64 MiB) is L2-resident
// (192 MB L2); output stores are non-temporal so they don't evict it.
// HBM traffic ~128 MiB -> ~5.5 us roofline at 23.3 TB/s.
//
// Wave-uniform values (partition id, warm-up count, batch base) are forced
// into SGPRs via readfirstlane so all control flow is scalar branches and
// EXEC stays all-1s through the WMMAs (ISA 7.12 requirement). Pointwise uses
// raw v_log_f32 / v_exp_f32 (branch-free, ~8 VALU ops per element).
// ---------------------------------------------------------------------------

#define SMOOTH  0.04f
#define QDECAY  0.96f
#define FLOORV  1e-6f

#define Bdim 8
#define Tdim 16384
#define Cdim 128

#define CT   16                 // channels per tile (WMMA N)
#define NCT  (Cdim / CT)        // 8 channel tiles
#define PART 32                 // partitions along T
#define LPART (Tdim / PART)     // 512 steps per partition
#define WARM 512                // warm-up halo (q^512 ~ 9e-10)
#define WAVES_PER_BLOCK 8       // 256 threads/block = 8 wave32

typedef __attribute__((ext_vector_type(2))) float v2f;
typedef __attribute__((ext_vector_type(8))) float v8f;

// fast pow for strictly-positive, non-denormal inputs:
// v^e = exp2(e * log2(v)) with 1-instruction hardware transcendentals
__device__ __forceinline__ float fast_pow_pos(float v, float e) {
    return __builtin_amdgcn_exp2f(e * __builtin_amdgcn_logf(v));
}

__global__ __launch_bounds__(256)
void pcen_wmma_scan(const float* __restrict__ x,
                    const float* __restrict__ alpha,
                    const float* __restrict__ delta,
                    const float* __restrict__ root,
                    float* __restrict__ out)
{
    const int lane = threadIdx.x & 31;
    const int wave = threadIdx.x >> 5;
    // wave-uniform by construction; readfirstlane makes that visible to the
    // compiler -> scalar branches / SALU addressing, EXEC never modified.
    const int unit = __builtin_amdgcn_readfirstlane(
                         (int)blockIdx.x * WAVES_PER_BLOCK + wave);  // 0..2047

    const int p     = unit & (PART - 1);
    const int ctile = (unit / PART) & (NCT - 1);
    const int b     = unit / (PART * NCT);

    const int n   = lane & 15;     // channel within tile / matrix row M
    const int khi = lane >> 4;     // half-wave selector
    const int c   = ctile * CT + n;

    // powers of q, q^0 .. q^16
    float qp[17];
    qp[0] = 1.0f;
#pragma unroll
    for (int i = 1; i <= 16; ++i) qp[i] = qp[i - 1] * QDECAY;

    // A operand (constant lower-triangular scan matrix), 4 K-chunks of 4.
    // ISA layout A 16x4 f32: lanes 0-15 -> M=lane, VGPR0=K0, VGPR1=K1;
    //                        lanes 16-31 -> M=lane-16, VGPR0=K2, VGPR1=K3.
    v2f Achunk[4];
    const int m = n;  // A-matrix row
#pragma unroll
    for (int cc = 0; cc < 4; ++cc) {
        const int k0 = 4 * cc + 2 * khi;
        Achunk[cc].x = (k0     <= m) ? SMOOTH * qp[m - k0]     : 0.0f;
        Achunk[cc].y = (k0 + 1 <= m) ? SMOOTH * qp[m - k0 - 1] : 0.0f;
    }

    // carry weights q^(M+1); C/D element v has M = v + 8*khi
    float cw[8];
#pragma unroll
    for (int v = 0; v < 8; ++v) cw[v] = qp[v + 8 * khi + 1];

    // per-channel PCEN params (loaded once)
    const float a_n  = fminf(alpha[c], 1.0f);
    const float r_n  = fmaxf(root[c], 1.0f);
    const float oor  = 1.0f / r_n;
    const float d_n  = delta[c];
    const float bias = fast_pow_pos(d_n, oor);

    const float* xb = x   + (size_t)b * Tdim * Cdim;
    float*       ob = out + (size_t)b * Tdim * Cdim;

    float carry;
    int t0, warm_tiles;
    if (p == 0) {                   // scalar branch
        carry = xb[c];              // y[-1] := x[0]  =>  y[0] = x[0]
        t0 = 0;
        warm_tiles = 0;
    } else {
        carry = 0.0f;
        t0 = p * LPART - WARM;
        warm_tiles = WARM / 16;
    }
    const int total_tiles = warm_tiles + LPART / 16;

    for (int tile = 0; tile < total_tiles; ++tile) {
        const int tt = t0 + tile * 16;

        // prefetch the next time-tile into near caches (global_prefetch_b8)
        if (tile + 1 < total_tiles)
            __builtin_prefetch(xb + (size_t)(tt + 16) * Cdim + c, 0, 3);

        // C accumulator pre-loaded with the carry term  q^(M+1) * carry
        v8f acc;
#pragma unroll
        for (int v = 0; v < 8; ++v) acc[v] = cw[v] * carry;

        // 4 chained f32 WMMAs: Y += A[:,k0:k0+4] * V[k0:k0+4,:]
        // B 4x16 f32 layout: VGPR0 lanes0-15=row k0, lanes16-31=row k0+2;
        //                    VGPR1 lanes0-15=row k0+1, lanes16-31=row k0+3.
#pragma unroll
        for (int cc = 0; cc < 4; ++cc) {
            const int r0 = tt + 4 * cc + 2 * khi;
            v2f bb;
            bb.x = xb[(size_t)(r0    ) * Cdim + c];
            bb.y = xb[(size_t)(r0 + 1) * Cdim + c];
            acc = __builtin_amdgcn_wmma_f32_16x16x4_f32(
                false, Achunk[cc], false, bb, (short)0, acc, false, false);
        }

        // new carry = Y[15][n] (VGPR7, lanes 16-31) broadcast to both halves
        carry = __shfl(acc[7], 16 + n, 32);

        if (tile >= warm_tiles) {   // scalar compare (warm_tiles in SGPR)
#pragma unroll
            for (int v = 0; v < 8; ++v) {
                const int t = tt + v + 8 * khi;        // C/D element M index
                const size_t idx = (size_t)t * Cdim + c;
                const float xv = xb[idx];              // L0/L2 hit (just read)
                const float y  = acc[v];
                // x / (FLOOR+y)^a  ==  x * exp2(-a * log2(FLOOR+y))
                const float dinv  = __builtin_amdgcn_exp2f(
                                        -a_n * __builtin_amdgcn_logf(FLOORV + y));
                const float inner = __builtin_fmaf(xv, dinv, d_n);   // >= delta
                // write-once output: non-temporal store, keep L2 for x
                __builtin_nontemporal_store(fast_pow_pos(inner, oor) - bias,
                                            &ob[idx]);
            }
        }
    }
}

extern "C" void kernel_launch(void* const* d_in, const int* in_sizes, int n_in,
                              void* d_out, int out_size, void* d_ws, size_t ws_size,
                              hipStream_t stream) {
    (void)in_sizes; (void)n_in; (void)out_size; (void)d_ws; (void)ws_size;
    const float* x     = (const float*)d_in[0];
    const float* alpha = (const float*)d_in[1];
    const float* delta = (const float*)d_in[2];
    const float* root  = (const float*)d_in[3];
    float* out = (float*)d_out;

    const int units = Bdim * NCT * PART;            // 2048 waves
    dim3 grid(units / WAVES_PER_BLOCK);             // 256 blocks
    dim3 block(32 * WAVES_PER_BLOCK);               // 256 threads (8 wave32)
    hipLaunchKernelGGL(pcen_wmma_scan, grid, block, 0, stream,
                       x, alpha, delta, root, out);
}